// QueryGNN_22076131901460
// MI455X (gfx1250) — compile-verified
//
#include <hip/hip_runtime.h>
#include <hip/hip_bf16.h>

typedef __attribute__((ext_vector_type(2))) float v2f;
typedef __attribute__((ext_vector_type(8))) float v8f;

#define LN_EPS 1e-5f

// ---------------------------------------------------------------------------
// 1) weighted degree: deg[row[e]] += ew[e]   (hardware f32 atomic, L2-resident)
// ---------------------------------------------------------------------------
__global__ __launch_bounds__(256) void deg_kernel(const int* __restrict__ row,
                                                  const float* __restrict__ ew,
                                                  float* __restrict__ deg, int E) {
  int e = blockIdx.x * blockDim.x + threadIdx.x;
  if (e < E) unsafeAtomicAdd(&deg[row[e]], ew[e]);
}

// deg -> deg^{-1/2}, 0 where deg==0 (in place)
__global__ __launch_bounds__(256) void dis_kernel(float* __restrict__ deg, int N) {
  int n = blockIdx.x * blockDim.x + threadIdx.x;
  if (n < N) {
    float d = deg[n];
    deg[n] = (d > 0.0f) ? rsqrtf(d) : 0.0f;
  }
}

// ---------------------------------------------------------------------------
// 2) GEMM + bias with V_WMMA_F32_16X16X4_F32.
//    One 16x16 output tile per wave; 8 waves per block cover 16 rows x 128 cols.
//    A fragment (16x4 f32, ISA 7.12.2): lanes 0-15 VGPR0/1 = K0/K1,
//    lanes 16-31 = K2/K3. B fragment symmetric (row striped across lanes).
//    C/D: VGPR v, lane l -> row = v + 8*(l/16), col = l%16.
// ---------------------------------------------------------------------------
__global__ __launch_bounds__(256)
void gemm_bias_wmma(const float* __restrict__ A, const float* __restrict__ B,
                    const float* __restrict__ bias, float* __restrict__ C,
                    int M, int K, int N) {
  const int lane = threadIdx.x & 31;
  const int wave = threadIdx.x >> 5;
  const int m0   = blockIdx.x * 16;
  const int n0   = (blockIdx.y * 8 + wave) * 16;
  if (n0 >= N) return;                      // wave-uniform exit (EXEC stays full)

  const int mrow = lane & 15;               // A row within tile
  const int half = lane >> 4;               // 0: K0/K1, 1: K2/K3
  const int kb   = half * 2;
  const int ncol = lane & 15;               // B/C column within tile

  int ar = m0 + mrow;  if (ar > M - 1) ar = M - 1;   // clamp tail reads
  const float* Arow = A + (size_t)ar * K;

  v8f acc = {};
  for (int k = 0; k < K; k += 4) {
    v2f a, b;
    a.x = Arow[k + kb];
    a.y = Arow[k + kb + 1];
    b.x = B[(size_t)(k + kb)     * N + n0 + ncol];
    b.y = B[(size_t)(k + kb + 1) * N + n0 + ncol];
    acc = __builtin_amdgcn_wmma_f32_16x16x4_f32(
        /*neg_a=*/false, a, /*neg_b=*/false, b,
        /*c_mod=*/(short)0, acc, /*reuse_a=*/false, /*reuse_b=*/false);
  }

  const float bv = bias[n0 + ncol];
#pragma unroll
  for (int v = 0; v < 8; ++v) {
    int r = m0 + v + 8 * half;
    if (r < M) C[(size_t)r * N + n0 + ncol] = acc[v] + bv;
  }
}

// ---------------------------------------------------------------------------
// 3) edge aggregation: out[row] += dis[row]*ew*dis[col] * H[col, :]
//    one wave per edge, lanes stride the feature dimension (coalesced gather,
//    f32 hardware atomics for the scatter; working set lives in 192MB L2).
// ---------------------------------------------------------------------------
template <int D>
__global__ __launch_bounds__(256)
void agg_kernel(const int* __restrict__ row, const int* __restrict__ col,
                const float* __restrict__ ew, const float* __restrict__ dis,
                const float* __restrict__ H, float* __restrict__ out, int E) {
  const int lane = threadIdx.x & 31;
  const int wave = threadIdx.x >> 5;
  const int e = blockIdx.x * 8 + wave;
  if (e >= E) return;
  const int r = row[e];
  const int c = col[e];
  const float norm = dis[r] * ew[e] * dis[c];
  const float* __restrict__ src = H + (size_t)c * D;
  float* __restrict__ dst = out + (size_t)r * D;
#pragma unroll
  for (int j = 0; j < D / 32; ++j) {
    int f = lane + 32 * j;
    unsafeAtomicAdd(&dst[f], norm * src[f]);
  }
}

// ---------------------------------------------------------------------------
// 4) LayerNorm (+ optional ReLU): one wave32 per row, shuffle reductions.
// ---------------------------------------------------------------------------
template <int D, bool RELU>
__global__ __launch_bounds__(256)
void ln_kernel(const float* __restrict__ X, const float* __restrict__ g,
               const float* __restrict__ b, float* __restrict__ Y, int N) {
  const int lane = threadIdx.x & 31;
  const int wave = threadIdx.x >> 5;
  const int n = blockIdx.x * 8 + wave;
  if (n >= N) return;
  constexpr int CPL = D / 32;
  const float* x = X + (size_t)n * D;

  float v[CPL];
  float s = 0.0f;
#pragma unroll
  for (int j = 0; j < CPL; ++j) { v[j] = x[lane + 32 * j]; s += v[j]; }
#pragma unroll
  for (int off = 16; off > 0; off >>= 1) s += __shfl_xor(s, off, 32);
  const float mean = s * (1.0f / D);

  float q = 0.0f;
#pragma unroll
  for (int j = 0; j < CPL; ++j) { float d = v[j] - mean; q += d * d; }
#pragma unroll
  for (int off = 16; off > 0; off >>= 1) q += __shfl_xor(q, off, 32);
  const float inv = rsqrtf(q * (1.0f / D) + LN_EPS);

  float* y = Y + (size_t)n * D;
#pragma unroll
  for (int j = 0; j < CPL; ++j) {
    int f = lane + 32 * j;
    float o = (v[j] - mean) * inv * g[f] + b[f];
    if (RELU) o = fmaxf(o, 0.0f);
    y[f] = o;
  }
}

// ---------------------------------------------------------------------------
// launch
// ---------------------------------------------------------------------------
extern "C" void kernel_launch(void* const* d_in, const int* in_sizes, int n_in,
                              void* d_out, int out_size, void* d_ws, size_t ws_size,
                              hipStream_t stream) {
  constexpr int D_IN = 384, D_H = 256, D_OUT = 128;

  const float* x   = (const float*)d_in[0];
  const int*   ei  = (const int*)d_in[1];     // [2, E] flat; int32 per harness
  const float* ew  = (const float*)d_in[2];
  const float* W1  = (const float*)d_in[3];
  const float* b1  = (const float*)d_in[4];
  const float* W2  = (const float*)d_in[5];
  const float* b2  = (const float*)d_in[6];
  const float* g1  = (const float*)d_in[7];
  const float* be1 = (const float*)d_in[8];
  const float* g2  = (const float*)d_in[9];
  const float* be2 = (const float*)d_in[10];
  float* out = (float*)d_out;

  const int N = in_sizes[0] / D_IN;           // 50000
  const int E = in_sizes[2];                  // 800000
  const int* row = ei;
  const int* col = ei + E;

  // workspace carve-up (floats)
  float* ws   = (float*)d_ws;
  float* dis  = ws;                                   // N
  float* buf1 = dis  + N;                             // N*256 (H1, later A2)
  float* buf2 = buf1 + (size_t)N * D_H;               // N*256 (A1 -> ln -> input of GEMM2)
  float* buf3 = buf2 + (size_t)N * D_H;               // N*128 (H2)

  const int mtiles = (N + 15) / 16;

  // degree -> dis
  hipMemsetAsync(dis, 0, (size_t)N * sizeof(float), stream);
  deg_kernel<<<(E + 255) / 256, 256, 0, stream>>>(row, ew, dis, E);
  dis_kernel<<<(N + 255) / 256, 256, 0, stream>>>(dis, N);

  // layer 1: H1 = x@W1 + b1
  gemm_bias_wmma<<<dim3(mtiles, D_H / 128), 256, 0, stream>>>(x, W1, b1, buf1, N, D_IN, D_H);
  // A1 = scatter(norm * H1[col])
  hipMemsetAsync(buf2, 0, (size_t)N * D_H * sizeof(float), stream);
  agg_kernel<D_H><<<(E + 7) / 8, 256, 0, stream>>>(row, col, ew, dis, buf1, buf2, E);
  // LN + ReLU in place
  ln_kernel<D_H, true><<<(N + 7) / 8, 256, 0, stream>>>(buf2, g1, be1, buf2, N);

  // layer 2: H2 = A1@W2 + b2
  gemm_bias_wmma<<<dim3(mtiles, 1), 256, 0, stream>>>(buf2, W2, b2, buf3, N, D_H, D_OUT);
  // A2 = scatter(norm * H2[col])  (reuse buf1)
  hipMemsetAsync(buf1, 0, (size_t)N * D_OUT * sizeof(float), stream);
  agg_kernel<D_OUT><<<(E + 7) / 8, 256, 0, stream>>>(row, col, ew, dis, buf3, buf1, E);
  // final LN -> d_out
  ln_kernel<D_OUT, false><<<(N + 7) / 8, 256, 0, stream>>>(buf1, g2, be2, out, N);
}